// GCN_18657337933853
// MI455X (gfx1250) — compile-verified
//
#include <hip/hip_runtime.h>
#include <hip/hip_bf16.h>
#include <math.h>

// ---------------- problem constants ----------------
#define N_NODES   100000
#define N_EDGES   1600000
#define D_FEAT    128
#define HIDDEN    128
#define N_LAYERS  3
#define N_GRAPHS  256
#define INV_TEMP  2.0f      // 1/0.5
#define NEG_W     0.8f

typedef __attribute__((ext_vector_type(2))) float v2f;
typedef __attribute__((ext_vector_type(8))) float v8f;

// ---------------- degree / norm ----------------
__global__ __launch_bounds__(256) void deg_kernel(const int* __restrict__ dst,
                                                  unsigned* __restrict__ deg, int ne) {
    int e = blockIdx.x * 256 + threadIdx.x;
    if (e < ne) atomicAdd(&deg[dst[e]], 1u);
}

__global__ __launch_bounds__(256) void rsqrt_kernel(const unsigned* __restrict__ deg,
                                                    float* __restrict__ dis, int n) {
    int i = blockIdx.x * 256 + threadIdx.x;
    if (i < n) dis[i] = rsqrtf((float)deg[i] + 1.0f);
}

// ---------------- dense GEMM: H[N,128] = X[N,128] @ W[128,128] ----------------
// One wave computes one 16x16 output tile with V_WMMA_F32_16X16X4_F32, K=128.
// Block: 256 threads = 8 waves -> 8 N-tiles (full 128-wide row of tiles).
// Grid: N/16 blocks (100000/16 = 6250, exact). EXEC is all-1s everywhere.
// The shared 16x128 A-strip of X is staged once into LDS (coalesced), padded
// to a 132-float row stride so all 32 lanes of a wave read distinct bank
// pairs when fetching their A fragments.
#define XS_STRIDE 132
__global__ __launch_bounds__(256) void gemm128_wmma(const float* __restrict__ X,
                                                    const float* __restrict__ W,
                                                    float* __restrict__ H) {
    __shared__ float xs[16 * XS_STRIDE];
    const int tid  = threadIdx.x;
    const int lane = tid & 31;
    const int wave = tid >> 5;
    const int m0   = blockIdx.x * 16;
    const int n0   = wave * 16;
    const int half = lane >> 4;           // 0: lanes 0-15, 1: lanes 16-31
    const int l15  = lane & 15;
    const int kb   = half * 2;            // K sub-offset 0 or 2

    // Stage X[m0..m0+15][0..127] (one contiguous 8KB block) into padded LDS.
    const float* xsrc = X + (size_t)m0 * D_FEAT;
#pragma unroll
    for (int q = 0; q < 8; ++q) {
        int i = tid + q * 256;            // 0..2047, coalesced
        xs[(i >> 7) * XS_STRIDE + (i & 127)] = xsrc[i];
    }
    __syncthreads();

    const float* arow = xs + l15 * XS_STRIDE;
    v8f acc = {};
#pragma unroll
    for (int k0 = 0; k0 < D_FEAT; k0 += 4) {
        v2f a, b;
        a.x = arow[k0 + kb];
        a.y = arow[k0 + kb + 1];
        b.x = W[(size_t)(k0 + kb) * HIDDEN + n0 + l15];
        b.y = W[(size_t)(k0 + kb + 1) * HIDDEN + n0 + l15];
        acc = __builtin_amdgcn_wmma_f32_16x16x4_f32(false, a, false, b,
                                                    (short)0, acc, false, false);
    }
    // D layout: VGPR r -> row (r + 8*half), col = l15
#pragma unroll
    for (int r = 0; r < 8; ++r)
        H[(size_t)(m0 + r + 8 * half) * HIDDEN + n0 + l15] = acc[r];
}

// ---------------- edge gather/scatter: agg[dst] += h[src] * enorm ----------------
__global__ __launch_bounds__(256) void scatter_kernel(const float* __restrict__ h,
                                                      const int* __restrict__ src,
                                                      const int* __restrict__ dst,
                                                      const float* __restrict__ dis,
                                                      float* __restrict__ agg, int ne) {
    int t = blockIdx.x * 256 + threadIdx.x;
    int e = t >> 5;
    if (e >= ne) return;
    int lane = t & 31;
    int s = src[e], d = dst[e];
    float en = dis[s] * dis[d];
    const float4* hv = (const float4*)(h + (size_t)s * HIDDEN);
    float4 v = hv[lane];
    float* ap = agg + (size_t)d * HIDDEN + lane * 4;
    atomicAdd(ap + 0, v.x * en);
    atomicAdd(ap + 1, v.y * en);
    atomicAdd(ap + 2, v.z * en);
    atomicAdd(ap + 3, v.w * en);
}

// ---------------- x = relu(agg + h*snorm + b) ----------------
__global__ __launch_bounds__(256) void finish_kernel(const float* __restrict__ agg,
                                                     const float* __restrict__ h,
                                                     const float* __restrict__ dis,
                                                     const float* __restrict__ bias,
                                                     float* __restrict__ xout, int n) {
    int t = blockIdx.x * 256 + threadIdx.x;
    if (t >= n * HIDDEN) return;
    int i = t >> 7, f = t & 127;
    float sn = dis[i] * dis[i];
    xout[t] = fmaxf(agg[t] + h[t] * sn + bias[f], 0.0f);
}

// ---------------- global add pool ----------------
__global__ __launch_bounds__(256) void pool_kernel(const float* __restrict__ x,
                                                   const int* __restrict__ batch,
                                                   float* __restrict__ g, int n) {
    int t = blockIdx.x * 256 + threadIdx.x;
    int i = t >> 5;
    if (i >= n) return;
    int lane = t & 31;
    int b = batch[i];
    const float4* xv = (const float4*)(x + (size_t)i * HIDDEN);
    float4 v = xv[lane];
    float* gp = g + (size_t)b * HIDDEN + lane * 4;
    atomicAdd(gp + 0, v.x);
    atomicAdd(gp + 1, v.y);
    atomicAdd(gp + 2, v.z);
    atomicAdd(gp + 3, v.w);
}

// ---------------- head: z1 = relu(concat(sc_g, fc_g) @ fc1_W + fc1_b) ----------------
__global__ __launch_bounds__(128) void head1_kernel(const float* __restrict__ scg,
                                                    const float* __restrict__ fcg,
                                                    const float* __restrict__ W1,
                                                    const float* __restrict__ b1,
                                                    float* __restrict__ z1) {
    __shared__ float row[2 * HIDDEN];
    int i = blockIdx.x, j = threadIdx.x;
    row[j]          = scg[i * HIDDEN + j];
    row[HIDDEN + j] = fcg[i * HIDDEN + j];
    __syncthreads();
    float s = b1[j];
#pragma unroll 4
    for (int k = 0; k < 2 * HIDDEN; ++k) s += row[k] * W1[k * HIDDEN + j];
    z1[i * HIDDEN + j] = fmaxf(s, 0.0f);
}

// ---------------- head: logp = log_softmax(z1 @ fc2_W + fc2_b) ----------------
__global__ __launch_bounds__(256) void head2_kernel(const float* __restrict__ z1,
                                                    const float* __restrict__ W2,
                                                    const float* __restrict__ b2,
                                                    float* __restrict__ logp) {
    int i = threadIdx.x;   // 256 graphs
    float y0 = b2[0], y1 = b2[1];
    const float* zr = z1 + (size_t)i * HIDDEN;
#pragma unroll 4
    for (int k = 0; k < HIDDEN; ++k) {
        float z = zr[k];
        y0 += z * W2[k * 2 + 0];
        y1 += z * W2[k * 2 + 1];
    }
    float m   = fmaxf(y0, y1);
    float lse = m + logf(expf(y0 - m) + expf(y1 - m));
    logp[i * 2 + 0] = y0 - lse;
    logp[i * 2 + 1] = y1 - lse;
}

// ---------------- row L2-normalize (512 rows: 256 sc, 256 fc) ----------------
__global__ __launch_bounds__(128) void norm_kernel(const float* __restrict__ scg,
                                                   const float* __restrict__ fcg,
                                                   float* __restrict__ scn,
                                                   float* __restrict__ fcn) {
    int b = blockIdx.x, t = threadIdx.x;
    const float* src = (b < N_GRAPHS) ? scg + (size_t)b * HIDDEN
                                      : fcg + (size_t)(b - N_GRAPHS) * HIDDEN;
    float* dst = (b < N_GRAPHS) ? scn + (size_t)b * HIDDEN
                                : fcn + (size_t)(b - N_GRAPHS) * HIDDEN;
    __shared__ float red[128];
    float v = src[t];
    red[t] = v * v;
    __syncthreads();
    for (int s = 64; s > 0; s >>= 1) {
        if (t < s) red[t] += red[t + s];
        __syncthreads();
    }
    float inv = 1.0f / fmaxf(sqrtf(red[0]), 1e-12f);
    dst[t] = v * inv;
}

// ---------------- contrastive losses ----------------
// Block i, threads j=0..255: logits row i of [pos | NEG_W*neg(offdiag)] (512 cols),
// log-softmax, pick diagonal. Accumulate mean into loss[0]/loss[1].
__global__ __launch_bounds__(256) void contrast_kernel(const float* __restrict__ scn,
                                                       const float* __restrict__ fcn,
                                                       float* __restrict__ loss) {
    int i = blockIdx.x, j = threadIdx.x;
    __shared__ float sci[HIDDEN], fci[HIDDEN];
    __shared__ float red[256];
    __shared__ float diag[2];
    if (j < HIDDEN) sci[j] = scn[(size_t)i * HIDDEN + j];
    else            fci[j - HIDDEN] = fcn[(size_t)i * HIDDEN + (j - HIDDEN)];
    __syncthreads();

    const float* fr = fcn + (size_t)j * HIDDEN;
    const float* sr = scn + (size_t)j * HIDDEN;
    float a_sf = 0.f, a_ss = 0.f, a_fs = 0.f, a_ff = 0.f;
#pragma unroll 4
    for (int k = 0; k < HIDDEN; ++k) {
        float fk = fr[k], sk = sr[k];
        a_sf += sci[k] * fk;
        a_ss += sci[k] * sk;
        a_fs += fci[k] * sk;
        a_ff += fci[k] * fk;
    }
    float L1 = a_sf * INV_TEMP;
    float L2 = (j == i) ? 0.0f : a_ss * INV_TEMP * NEG_W;
    float M1 = a_fs * INV_TEMP;
    float M2 = (j == i) ? 0.0f : a_ff * INV_TEMP * NEG_W;
    if (j == i) { diag[0] = L1; diag[1] = M1; }

    // sc row: max
    red[j] = fmaxf(L1, L2); __syncthreads();
    for (int s = 128; s > 0; s >>= 1) { if (j < s) red[j] = fmaxf(red[j], red[j + s]); __syncthreads(); }
    float msc = red[0]; __syncthreads();
    // sc row: sumexp
    red[j] = expf(L1 - msc) + expf(L2 - msc); __syncthreads();
    for (int s = 128; s > 0; s >>= 1) { if (j < s) red[j] += red[j + s]; __syncthreads(); }
    float Ssc = red[0]; __syncthreads();
    // fc row: max
    red[j] = fmaxf(M1, M2); __syncthreads();
    for (int s = 128; s > 0; s >>= 1) { if (j < s) red[j] = fmaxf(red[j], red[j + s]); __syncthreads(); }
    float mfc = red[0]; __syncthreads();
    // fc row: sumexp
    red[j] = expf(M1 - mfc) + expf(M2 - mfc); __syncthreads();
    for (int s = 128; s > 0; s >>= 1) { if (j < s) red[j] += red[j + s]; __syncthreads(); }
    float Sfc = red[0];

    if (j == 0) {
        float li = -(diag[0] - msc - logf(Ssc));
        float lt = -(diag[1] - mfc - logf(Sfc));
        atomicAdd(&loss[0], li * (1.0f / N_GRAPHS));
        atomicAdd(&loss[1], lt * (1.0f / N_GRAPHS));
    }
}

__global__ __launch_bounds__(512) void final_kernel(const float* __restrict__ logp,
                                                    const float* __restrict__ loss,
                                                    float* __restrict__ out) {
    int t = threadIdx.x;   // 512
    out[t] = logp[t] + 0.5f * (loss[0] + loss[1]);
}

// ---------------- launcher ----------------
extern "C" void kernel_launch(void* const* d_in, const int* in_sizes, int n_in,
                              void* d_out, int out_size, void* d_ws, size_t ws_size,
                              hipStream_t stream) {
    const float* sc_x   = (const float*)d_in[0];
    const float* fc_x   = (const float*)d_in[1];
    const int*   sc_ei  = (const int*)d_in[2];
    const int*   fc_ei  = (const int*)d_in[3];
    const int*   batch  = (const int*)d_in[4];
    const float* sc_W   = (const float*)d_in[5];
    const float* sc_b   = (const float*)d_in[6];
    const float* fc_W   = (const float*)d_in[7];
    const float* fc_b   = (const float*)d_in[8];
    const float* fc1_W  = (const float*)d_in[9];
    const float* fc1_b  = (const float*)d_in[10];
    const float* fc2_W  = (const float*)d_in[11];
    const float* fc2_b  = (const float*)d_in[12];
    float* out = (float*)d_out;

    // bump allocator over workspace
    char* wsp = (char*)d_ws;
    auto alloc = [&](size_t bytes) -> void* {
        void* p = (void*)wsp;
        wsp += (bytes + 255) & ~(size_t)255;
        return p;
    };
    const size_t NF = (size_t)N_NODES * HIDDEN;
    float*    x_cur = (float*)alloc(NF * 4);
    float*    hbuf  = (float*)alloc(NF * 4);
    float*    agg   = (float*)alloc(NF * 4);
    unsigned* deg   = (unsigned*)alloc((size_t)N_NODES * 4);
    float*    dis   = (float*)alloc((size_t)N_NODES * 4);
    float*    sc_g  = (float*)alloc((size_t)N_GRAPHS * HIDDEN * 4);
    float*    fc_g  = (float*)alloc((size_t)N_GRAPHS * HIDDEN * 4);
    float*    scn   = (float*)alloc((size_t)N_GRAPHS * HIDDEN * 4);
    float*    fcn   = (float*)alloc((size_t)N_GRAPHS * HIDDEN * 4);
    float*    z1    = (float*)alloc((size_t)N_GRAPHS * HIDDEN * 4);
    float*    logp  = (float*)alloc((size_t)N_GRAPHS * 2 * 4);
    float*    lossb = (float*)alloc(2 * 4);

    const int gE   = (N_EDGES + 255) / 256;
    const int gN   = (N_NODES + 255) / 256;
    const int gNF  = (int)((NF + 255) / 256);
    const int gE32 = (int)(((size_t)N_EDGES * 32 + 255) / 256);
    const int gN32 = (int)(((size_t)N_NODES * 32 + 255) / 256);
    const int gGEMM = N_NODES / 16;   // 6250, exact

    // ---- two GCN branches ----
    for (int br = 0; br < 2; ++br) {
        const float* x0   = br == 0 ? sc_x : fc_x;
        const int*   ei   = br == 0 ? sc_ei : fc_ei;
        const float* Wall = br == 0 ? sc_W : fc_W;
        const float* ball = br == 0 ? sc_b : fc_b;
        float*       g    = br == 0 ? sc_g : fc_g;
        const int* esrc = ei;
        const int* edst = ei + N_EDGES;

        hipMemsetAsync(deg, 0, (size_t)N_NODES * 4, stream);
        deg_kernel<<<gE, 256, 0, stream>>>(edst, deg, N_EDGES);
        rsqrt_kernel<<<gN, 256, 0, stream>>>(deg, dis, N_NODES);

        const float* xin = x0;
        for (int l = 0; l < N_LAYERS; ++l) {
            gemm128_wmma<<<gGEMM, 256, 0, stream>>>(xin, Wall + (size_t)l * D_FEAT * HIDDEN, hbuf);
            hipMemsetAsync(agg, 0, NF * 4, stream);
            scatter_kernel<<<gE32, 256, 0, stream>>>(hbuf, esrc, edst, dis, agg, N_EDGES);
            finish_kernel<<<gNF, 256, 0, stream>>>(agg, hbuf, dis, ball + (size_t)l * HIDDEN,
                                                   x_cur, N_NODES);
            xin = x_cur;
        }
        hipMemsetAsync(g, 0, (size_t)N_GRAPHS * HIDDEN * 4, stream);
        pool_kernel<<<gN32, 256, 0, stream>>>(x_cur, batch, g, N_NODES);
    }

    // ---- classifier head ----
    head1_kernel<<<N_GRAPHS, 128, 0, stream>>>(sc_g, fc_g, fc1_W, fc1_b, z1);
    head2_kernel<<<1, 256, 0, stream>>>(z1, fc2_W, fc2_b, logp);

    // ---- contrastive block ----
    norm_kernel<<<2 * N_GRAPHS, 128, 0, stream>>>(sc_g, fc_g, scn, fcn);
    hipMemsetAsync(lossb, 0, 2 * 4, stream);
    contrast_kernel<<<N_GRAPHS, 256, 0, stream>>>(scn, fcn, lossb);

    final_kernel<<<1, 512, 0, stream>>>(logp, lossb, out);
    (void)in_sizes; (void)n_in; (void)out_size; (void)ws_size;
}